// RQSplineHead_7928509628447
// MI455X (gfx1250) — compile-verified
//
#include <hip/hip_runtime.h>

typedef __attribute__((ext_vector_type(2))) float v2f;
typedef __attribute__((ext_vector_type(4))) float v4f;
typedef __attribute__((ext_vector_type(8))) float v8f;

#define IN_DIM   256
#define OUT_DIM  27
#define PAD_N    32     // padded output cols (2 WMMA N-tiles)
#define NKNOT    9      // K1 = N_KNOTS + 1
#define NBINS    128
#define ROWS_PB  128    // rows per block (8 waves x 16 rows)

__device__ __forceinline__ float softplus_eps(float z) {
    // stable softplus: max(z,0) + log1p(exp(-|z|)), then + 1e-4
    return fmaxf(z, 0.0f) + log1pf(expf(-fabsf(z))) + 1.0e-4f;
}

// Pad W (27x256) -> Wp (32x256, zero rows 27..31), bias (27) -> bp (32, zero tail).
__global__ __launch_bounds__(256) void rqspline_prep(
    const float* __restrict__ W, const float* __restrict__ bias,
    float* __restrict__ Wp, float* __restrict__ bp)
{
    const int idx = blockIdx.x * 256 + threadIdx.x;       // 0 .. 32*256-1
    const int n = idx >> 8;
    Wp[idx] = (n < OUT_DIM) ? W[idx - (32 - OUT_DIM) * 0 - (n - n) + (n * IN_DIM + (idx & 255)) - idx + idx] : 0.0f;
    // (simple form below; keep it readable)
    Wp[idx] = (n < OUT_DIM) ? W[n * IN_DIM + (idx & 255)] : 0.0f;
    if (idx < PAD_N) bp[idx] = (idx < OUT_DIM) ? bias[idx] : 0.0f;
}

__global__ __launch_bounds__(256) void rqspline_kernel(
    const float* __restrict__ h, const float* __restrict__ Wp,
    const float* __restrict__ bp, float* __restrict__ out, int Btot)
{
    __shared__ float sp[ROWS_PB][33];   // post-softplus params (27 used, 28..32 junk)
    __shared__ float sx[ROWS_PB][10];   // cumx knots
    __shared__ float sy[ROWS_PB][10];   // cumy knots
    __shared__ float sd[ROWS_PB][12];   // padded derivatives (11 used)

    const int tid  = threadIdx.x;
    const int wave = tid >> 5;
    const int lane = tid & 31;
    const int l16  = lane & 15;
    const int half = lane >> 4;

    const int rowbase = blockIdx.x * ROWS_PB + wave * 16;

    // ---------------- Stage 1: params = h @ W^T + b via V_WMMA_F32_16X16X4_F32
    v8f acc0 = {};   // output cols 0..15
    v8f acc1 = {};   // output cols 16..31

    const int n0 = l16;
    const int n1 = 16 + l16;

    int arow = rowbase + l16;
    if (arow >= Btot) arow = Btot - 1;              // clamp (B divisible anyway)
    const float* __restrict__ hrow = h  + (size_t)arow * IN_DIM;
    const float* __restrict__ w0p  = Wp + (size_t)n0 * IN_DIM;
    const float* __restrict__ w1p  = Wp + (size_t)n1 * IN_DIM;

    #pragma unroll 8
    for (int k0 = 0; k0 < IN_DIM; k0 += 4) {
        const int kk = k0 + 2 * half;               // ISA fragment layout:
        v2f a  = *(const v2f*)(hrow + kk);          //  lanes 0-15: K=k0,k0+1 ; lanes 16-31: K=k0+2,k0+3
        v2f b0 = *(const v2f*)(w0p  + kk);
        v2f b1 = *(const v2f*)(w1p  + kk);
        acc0 = __builtin_amdgcn_wmma_f32_16x16x4_f32(false, a, false, b0, (short)0, acc0, false, false);
        acc1 = __builtin_amdgcn_wmma_f32_16x16x4_f32(false, a, false, b1, (short)0, acc1, false, false);
    }

    // bias + softplus + scatter C fragments to LDS (unconditional; cols 27..31 unused)
    const float b0v = bp[n0];
    const float b1v = bp[n1];
    #pragma unroll
    for (int g = 0; g < 8; ++g) {
        const int rl = wave * 16 + g + 8 * half;    // C layout: VGPR g -> row g (lanes 0-15), row 8+g (lanes 16-31)
        sp[rl][n0] = softplus_eps(acc0[g] + b0v);
        sp[rl][n1] = softplus_eps(acc1[g] + b1v);
    }
    __syncthreads();

    // ---------------- Stage 2: per-row knot preparation (1 thread / row)
    if (tid < ROWS_PB) {
        const float* __restrict__ p = sp[tid];
        float sw = 0.0f, sh = 0.0f;
        #pragma unroll
        for (int i = 0; i < NKNOT; ++i) { sw += p[i]; sh += p[NKNOT + i]; }
        const float isw = 1.0f / sw;
        const float ish = 1.0f / sh;
        float cx = 0.0f, cy = 0.0f;
        sx[tid][0] = 0.0f; sy[tid][0] = 0.0f;
        #pragma unroll
        for (int i = 0; i < NKNOT; ++i) {
            cx += p[i]         * isw;  sx[tid][i + 1] = cx;
            cy += p[NKNOT + i] * ish;  sy[tid][i + 1] = cy;
        }
        sd[tid][0] = 1.0f;
        #pragma unroll
        for (int i = 0; i < NKNOT; ++i) sd[tid][1 + i] = p[2 * NKNOT + i];
        sd[tid][NKNOT + 1] = 1.0f;
    }
    __syncthreads();

    // ---------------- Stage 3: spline CDF at 129 uniform edges, 2 threads / row
    const int row  = tid >> 1;
    const int hf   = tid & 1;
    const int rowg = blockIdx.x * ROWS_PB + row;
    if (rowg >= Btot) return;

    const float* __restrict__ SX = sx[row];
    const float* __restrict__ SY = sy[row];
    const float* __restrict__ SD = sd[row];
    float* __restrict__ op = out + (size_t)rowg * NBINS;

    int bin = 0;
    float cxb, cxn, cyb, cyn, db, dn;

    const int j0 = hf * (NBINS / 2);
    float x = (float)j0 * (1.0f / (float)NBINS);
    while (bin < NKNOT - 1 && SX[bin + 1] < x) ++bin;
    cxb = SX[bin]; cxn = SX[bin + 1];
    cyb = SY[bin]; cyn = SY[bin + 1];
    db  = SD[bin]; dn  = SD[bin + 1];

    auto evalF = [&](float xv) -> float {
        const float w     = cxn - cxb;
        const float hh    = cyn - cyb;
        const float delta = hh / w;
        const float theta = (xv - cxb) / w;
        const float t1m   = theta * (1.0f - theta);
        const float num   = hh * (delta * theta * theta + db * t1m);
        const float den   = delta + (db + dn - 2.0f * delta) * t1m;
        return cyb + num / den;
    };

    float Fprev = evalF(x);
    for (int jj = 0; jj < NBINS / 2; jj += 4) {
        v4f r;
        #pragma unroll
        for (int q = 0; q < 4; ++q) {
            const int j = j0 + jj + q + 1;
            x = (float)j * (1.0f / (float)NBINS);
            while (bin < NKNOT - 1 && cxn < x) {
                ++bin;
                cxb = SX[bin]; cxn = SX[bin + 1];
                cyb = SY[bin]; cyn = SY[bin + 1];
                db  = SD[bin]; dn  = SD[bin + 1];
            }
            const float F  = evalF(x);
            const float pr = F - Fprev;
            Fprev = F;
            r[q] = logf(fmaxf(pr, 1.0e-8f));
        }
        *(v4f*)(op + j0 + jj) = r;   // 16B-aligned: op is 512B-aligned, j0+jj multiple of 4
    }
}

extern "C" void kernel_launch(void* const* d_in, const int* in_sizes, int n_in,
                              void* d_out, int out_size, void* d_ws, size_t ws_size,
                              hipStream_t stream) {
    const float* h = (const float*)d_in[0];
    const float* W = (const float*)d_in[1];
    const float* b = (const float*)d_in[2];
    float* out = (float*)d_out;

    float* Wp = (float*)d_ws;                       // 32*256 floats
    float* bp = Wp + PAD_N * IN_DIM;                // 32 floats

    rqspline_prep<<<PAD_N * IN_DIM / 256, 256, 0, stream>>>(W, b, Wp, bp);

    const int Btot = in_sizes[0] / IN_DIM;           // 131072
    const int grid = (Btot + ROWS_PB - 1) / ROWS_PB; // 1024
    rqspline_kernel<<<grid, 256, 0, stream>>>(h, Wp, bp, out, Btot);
}